// GAPooling_53334903882144
// MI455X (gfx1250) — compile-verified
//
#include <hip/hip_runtime.h>

// GAPooling: out[b,n,c] = mean_k x[b, idx[b,n,k], c]
// B=16, N=4096, C=64, K=32 (shapes fixed by the reference; grid derived from in_sizes).
//
// One wave32 per point. Memory-bound gather: each x row is 256B, fetched as
// 64B-contiguous segments per half-wave. Final 4-way partial reduction + 1/32
// scaling done exactly in fp32 on the matrix unit via V_WMMA_F32_16X16X4_F32
// with B == 1/32 (ones-matmul reduction).

typedef __attribute__((ext_vector_type(2))) float v2f;
typedef __attribute__((ext_vector_type(8))) float v8f;

namespace {
constexpr int KNB   = 32;   // neighbors per point
constexpr int CCH   = 64;   // channels
constexpr int NLOG2 = 12;   // N = 4096
constexpr int WAVES_PER_BLOCK = 8;  // 256 threads, wave32
}

// Select d[r], r in [0,8): 3-level cndmask tree (no scratch, no divergence).
__device__ __forceinline__ float sel8(v8f d, int r) {
    float x01 = (r & 1) ? d[1] : d[0];
    float x23 = (r & 1) ? d[3] : d[2];
    float x45 = (r & 1) ? d[5] : d[4];
    float x67 = (r & 1) ? d[7] : d[6];
    float y03 = (r & 2) ? x23 : x01;
    float y47 = (r & 2) ? x67 : x45;
    return (r & 4) ? y47 : y03;
}

__device__ __forceinline__ v8f vsel8(bool c, v8f a, v8f b) {
    v8f r;
#pragma unroll
    for (int i = 0; i < 8; ++i) r[i] = c ? a[i] : b[i];
    return r;
}

__global__ __launch_bounds__(256) void gapool_wmma_kernel(
    const float* __restrict__ x,                 // [B, N, C] fp32
    const unsigned long long* __restrict__ idx,  // [B, N, K] int64
    float* __restrict__ out)                     // [B, N, C] fp32
{
    const int lane = threadIdx.x & 31;
    const int wave = threadIdx.x >> 5;
    const int p    = blockIdx.x * WAVES_PER_BLOCK + wave;  // flat point id = b*N + n

    // Per-batch base pointer (uniform per wave -> SGPR, loads use saddr+voffset).
    const float* __restrict__ xb = x + ((size_t)(p >> NLOG2) << NLOG2) * CCH;

    // One coalesced 256B load grabs all 32 neighbor indices (lane k -> idx[k]).
    // idx is streamed once: non-temporal so it doesn't evict x from L2.
    const unsigned long long i64 =
        __builtin_nontemporal_load(idx + (((size_t)p) << 5) + lane);
    const int jlane = (int)i64;  // indices < 4096, low word suffices

    const bool hi    = lane >= 16;
    const int  cbase = lane & 15;  // channel-in-group, matches A-matrix row M

    // A-matrix accumulators (16x4 f32 layout): lanes 0-15 hold K=0,1 slots,
    // lanes 16-31 hold K=2,3 slots, each slot = sum of 8 neighbors.
    v2f acc0 = {0.f, 0.f}, acc1 = {0.f, 0.f}, acc2 = {0.f, 0.f}, acc3 = {0.f, 0.f};

#pragma unroll
    for (int t = 0; t < 8; ++t) {
        // Half-wave broadcasts: lanes 0-15 take neighbors t / 8+t,
        // lanes 16-31 take neighbors 16+t / 24+t. Constant-lane readlanes.
        const int ja_lo = __builtin_amdgcn_readlane(jlane, t);
        const int ja_hi = __builtin_amdgcn_readlane(jlane, 16 + t);
        const int jb_lo = __builtin_amdgcn_readlane(jlane, 8 + t);
        const int jb_hi = __builtin_amdgcn_readlane(jlane, 24 + t);
        const int j0 = hi ? ja_hi : ja_lo;
        const int j1 = hi ? jb_hi : jb_lo;

        // Each load instruction: 64B contiguous from row j0 (lanes 0-15)
        // + 64B contiguous from row j_hi (lanes 16-31).
        const int off0 = (j0 << 6) + cbase;
        const int off1 = (j1 << 6) + cbase;
        acc0.x += xb[off0];      acc1.x += xb[off0 + 16];
        acc2.x += xb[off0 + 32]; acc3.x += xb[off0 + 48];
        acc0.y += xb[off1];      acc1.y += xb[off1 + 16];
        acc2.y += xb[off1 + 32]; acc3.y += xb[off1 + 48];
    }

    // D[m,n] = sum_k A[m,k] * (1/32): exact fp32 4-way reduce + mean scale on
    // the matrix unit. B is a constant matrix -> layout-independent.
    const v2f bscale = {1.0f / 32.0f, 1.0f / 32.0f};
    const v8f czero  = {};
    v8f d0 = __builtin_amdgcn_wmma_f32_16x16x4_f32(false, acc0, false, bscale,
                                                   (short)0, czero, false, false);
    v8f d1 = __builtin_amdgcn_wmma_f32_16x16x4_f32(false, acc1, false, bscale,
                                                   (short)0, czero, false, false);
    v8f d2 = __builtin_amdgcn_wmma_f32_16x16x4_f32(false, acc2, false, bscale,
                                                   (short)0, czero, false, false);
    v8f d3 = __builtin_amdgcn_wmma_f32_16x16x4_f32(false, acc3, false, bscale,
                                                   (short)0, czero, false, false);

    // D layout: VGPR r holds M = r + 8*(lane>=16), value replicated over N.
    // Each lane stores two distinct (m, group) channels; both store
    // instructions cover two contiguous 64B segments per wave (8-lane 32B pieces).
    const int  r    = lane & 7;
    const bool gsel = (lane >> 3) & 1;            // groups {2,3} vs {0,1}
    const int  mst  = (lane & 7) + (hi ? 8 : 0);  // M row this lane extracts
    const int  g1   = gsel ? 2 : 0;

    const v8f dA = vsel8(gsel, d2, d0);
    const v8f dB = vsel8(gsel, d3, d1);
    const float v1 = sel8(dA, r);
    const float v2 = sel8(dB, r);

    // Output streamed once: non-temporal stores keep x resident in L2.
    float* __restrict__ op = out + ((size_t)p << 6);
    __builtin_nontemporal_store(v1, op + mst + 16 * g1);
    __builtin_nontemporal_store(v2, op + mst + 16 * (g1 + 1));
}

extern "C" void kernel_launch(void* const* d_in, const int* in_sizes, int n_in,
                              void* d_out, int out_size, void* d_ws, size_t ws_size,
                              hipStream_t stream) {
    const float* x = (const float*)d_in[0];
    const unsigned long long* idx = (const unsigned long long*)d_in[1];  // int64
    float* out = (float*)d_out;

    const int npoints = in_sizes[0] / CCH;          // B*N = 65536
    const int blocks  = npoints / WAVES_PER_BLOCK;  // exact: 8192 full blocks
    gapool_wmma_kernel<<<blocks, 256, 0, stream>>>(x, idx, out);
}